// RoIPooling_89206470738596
// MI455X (gfx1250) — compile-verified
//
#include <hip/hip_runtime.h>

// ROI-Align (crop_and_resize, TF2 half-pixel bilinear) for MI455X / gfx1250.
//
// Vertical interpolation is expressed as D = A x B with V_WMMA_F32_16X16X4_F32:
//   M = pool rows (7 used of 16), N = 16 channels, K = 2*py + tap (14 used of 16)
// A = vertical weight matrix (per box), B = horizontally-lerped feature rows.
// Per the CDNA5 ISA 32-bit operand layouts, lane (half = lane>>4, chan = lane&15)
// supplies, for WMMA chunk c and fragment element v:
//   K = 4c + v + 2*half  ==>  py_src = 2c + half, tap = v
// so both A and B fragments are built with lane-local closed forms (no LDS,
// no cross-lane shuffles, no scratch arrays).

typedef __attribute__((ext_vector_type(2))) float v2f;
typedef __attribute__((ext_vector_type(8))) float v8f;

#if defined(__HIP_DEVICE_COMPILE__) && \
    !__has_builtin(__builtin_amdgcn_wmma_f32_16x16x4_f32)
#error "missing __builtin_amdgcn_wmma_f32_16x16x4_f32 on this toolchain"
#endif

#define POOLN 7
#define FH 256
#define FW 256
#define FC 512
#define NTILE 16           // WMMA N (channels per tile)
#define NCT (FC / NTILE)   // 32 channel tiles

__global__ __launch_bounds__(256) void roi_align_wmma_kernel(
    const float* __restrict__ fm,    // [FH][FW][FC]
    const int*   __restrict__ props, // [N][4] = x, y, w, h
    float*       __restrict__ out)   // [N][7][7][FC]
{
  const int box  = blockIdx.x;
  const int lane = threadIdx.x & 31;
  const int wid  = threadIdx.x >> 5;   // 0..7
  const int half = lane >> 4;          // 0: lanes 0-15, 1: lanes 16-31
  const int chan = lane & 15;          // channel-within-tile; also M row for A

  const int bx = props[box * 4 + 0];
  const int by = props[box * 4 + 1];
  const int bw = props[box * 4 + 2];
  const int bh = props[box * 4 + 3];

  // ---- vertical axis: this lane's 4 assigned pool rows (py = 2c + half) ----
  long  rlo[4], rhi[4];    // element offsets of source rows y0[py], y1[py]
  float wlo[4], whi[4];    // (1-fy), fy for those rows
  const float yscale = (float)bh / (float)POOLN;
  const float yhi    = fmaxf((float)bh - 1.0f, 0.0f);
#pragma unroll
  for (int c = 0; c < 4; ++c) {
    int  p     = 2 * c + half;         // pool row this lane feeds in chunk c
    bool valid = (p < POOLN);          // p == 7 (c==3, half==1) is K-padding
    float s = ((float)p + 0.5f) * yscale - 0.5f;
    s = fminf(fmaxf(s, 0.0f), yhi);
    int   i0 = (int)s;                 // s >= 0 -> trunc == floor
    int   i1 = min(i0 + 1, bh - 1);
    float fy = s - (float)i0;
    rlo[c] = valid ? (long)(by + i0) * (FW * FC) : 0;  // clamp pads to row 0
    rhi[c] = valid ? (long)(by + i1) * (FW * FC) : 0;
    wlo[c] = 1.0f - fy;
    whi[c] = fy;
  }

  // ---- A fragments (constant per box, reused for every channel tile/px) ----
  // A[m][k] nonzero only when m == py_src(k): row py's two vertical weights.
  v2f a[4];
#pragma unroll
  for (int c = 0; c < 4; ++c) {
    int  p   = 2 * c + half;
    bool sel = (p < POOLN) && (chan == p);
    a[c][0] = sel ? wlo[c] : 0.0f;     // tap 0 -> (1 - fy)
    a[c][1] = sel ? whi[c] : 0.0f;     // tap 1 -> fy
  }

  const float xscale = (float)bw / (float)POOLN;
  const float xhi    = fmaxf((float)bw - 1.0f, 0.0f);

#pragma unroll
  for (int t = 0; t < NCT / 8; ++t) {      // 4 channel tiles per wave
    const int  ct = wid + 8 * t;           // 0..31
    const long cb = (long)ct * NTILE + chan;
#pragma unroll
    for (int px = 0; px < POOLN; ++px) {
      // horizontal sample coords (fp32 math identical to reference)
      float s = ((float)px + 0.5f) * xscale - 0.5f;
      s = fminf(fmaxf(s, 0.0f), xhi);
      int   i0 = (int)s;
      int   i1 = min(i0 + 1, bw - 1);
      float fx = s - (float)i0;
      float gx = 1.0f - fx;
      const long c0 = (long)(bx + i0) * FC + cb;
      const long c1 = (long)(bx + i1) * FC + cb;

      // ---- B fragments: horizontal lerp of the rows this lane supplies ----
      v2f b[4];
#pragma unroll
      for (int c = 0; c < 4; ++c) {
        float p00 = fm[rlo[c] + c0];
        float p01 = fm[rlo[c] + c1];
        float p10 = fm[rhi[c] + c0];
        float p11 = fm[rhi[c] + c1];
        b[c][0] = p00 * gx + p01 * fx;   // K tap 0 (row y0[py])
        b[c][1] = p10 * gx + p11 * fx;   // K tap 1 (row y1[py])
      }

      // ---- vertical interpolation: 4 chained f32 WMMAs (K = 16 >= 14) ----
      v8f d = {};
#if defined(__HIP_DEVICE_COMPILE__)
      d = __builtin_amdgcn_wmma_f32_16x16x4_f32(false, a[0], false, b[0],
                                                (short)0, d, false, false);
      d = __builtin_amdgcn_wmma_f32_16x16x4_f32(false, a[1], false, b[1],
                                                (short)0, d, false, false);
      d = __builtin_amdgcn_wmma_f32_16x16x4_f32(false, a[2], false, b[2],
                                                (short)0, d, false, false);
      d = __builtin_amdgcn_wmma_f32_16x16x4_f32(false, a[3], false, b[3],
                                                (short)0, d, false, false);
#else
      // host pass: __global__ body is parsed but never executed; keep the
      // data flow alive without target-specific builtins.
      d[0] = a[0][0] + b[0][0];
#endif

      // D layout: VGPR j, lanes 0-15 hold (py = j, channel = chan).
      if (half == 0) {
#pragma unroll
        for (int py = 0; py < POOLN; ++py) {
          out[(((long)box * POOLN + py) * POOLN + px) * FC + cb] = d[py];
        }
      }
    }
  }
}

extern "C" void kernel_launch(void* const* d_in, const int* in_sizes, int n_in,
                              void* d_out, int out_size, void* d_ws, size_t ws_size,
                              hipStream_t stream) {
  const float* fm    = (const float*)d_in[0];
  const int*   props = (const int*)d_in[1];
  float*       out   = (float*)d_out;
  const int    nbox  = in_sizes[1] / 4;   // proposals are [N][4]
  roi_align_wmma_kernel<<<nbox, 256, 0, stream>>>(fm, props, out);
}